// GRCN_86449101734543
// MI455X (gfx1250) — compile-verified
//
#include <hip/hip_runtime.h>
#include <hip/hip_bf16.h>
#include <math.h>

#define NUM_USER  50000
#define NUM_ITEM  40000
#define NNODE     90000
#define DIM       64
#define NEG_SLOPE 0.01f

typedef __attribute__((ext_vector_type(2))) float v2f;
typedef __attribute__((ext_vector_type(8))) float v8f;

__device__ __forceinline__ float leaky(float v) { return v >= 0.f ? v : NEG_SLOPE * v; }
// monotone float <-> ordered-int mapping so atomicMax(int) implements float max
__device__ __forceinline__ int f2ord(float f) {
  int i = __float_as_int(f);
  return i >= 0 ? i : (i ^ 0x7fffffff);
}
__device__ __forceinline__ float ord2f(int m) {
  return __int_as_float(m >= 0 ? m : (m ^ 0x7fffffff));
}

// ---------------------------------------------------------------------------
// Dense projection via fp32 WMMA: out[M x 64] = leaky_relu(A[MxK] @ W[Kx64] + b)
// One wave computes a 16x64 output tile with 4 accumulators of 16x16.
// W is staged through LDS in 64-K slabs shared by the 8 waves of the block
// (8x less global W traffic; B-fragments come from ds_load).
// A-frag layout (ISA 7.12.2, 32-bit A 16x4): lane(l15,half) holds
//   A[row0+l15][k0+2*half], A[row0+l15][k0+2*half+1]
// B-frag (4x16, rows striped across lanes): vgpr v, lane -> B[k0+half+2v][n0+l15]
// D 16x16: vgpr r, lane -> D[r + 8*half][l15]
// ---------------------------------------------------------------------------
__global__ void k_proj_wmma(const float* __restrict__ A, const float* __restrict__ W,
                            const float* __restrict__ bias, float* __restrict__ out,
                            int M, int K) {
  __shared__ float Ws[64][DIM + 1];  // +1 pad to stagger banks
  int waveInBlk = threadIdx.x >> 5;  // 0..7
  int lane = threadIdx.x & 31;
  int row0 = (blockIdx.x * 8 + waveInBlk) * 16;
  int half = lane >> 4, l15 = lane & 15;
  bool active = row0 < M;

  v8f acc[4] = {};
  const float* arow = A + (size_t)(row0 + l15) * K;  // only read when active

  for (int kc = 0; kc < K; kc += 64) {
    __syncthreads();  // protect previous slab's readers
    for (int i = threadIdx.x; i < 64 * DIM; i += 256) {
      Ws[i >> 6][i & 63] = W[(size_t)(kc + (i >> 6)) * DIM + (i & 63)];
    }
    __syncthreads();
    if (active) {
      for (int k0 = 0; k0 < 64; k0 += 4) {
        v2f a;
        a.x = arow[kc + k0 + 2 * half];
        a.y = arow[kc + k0 + 2 * half + 1];
#pragma unroll
        for (int nt = 0; nt < 4; ++nt) {
          v2f b;
          b.x = Ws[k0 + half][nt * 16 + l15];
          b.y = Ws[k0 + half + 2][nt * 16 + l15];
          acc[nt] = __builtin_amdgcn_wmma_f32_16x16x4_f32(
              false, a, false, b, (short)0, acc[nt], false, false);
        }
      }
    }
  }
  if (!active) return;
#pragma unroll
  for (int nt = 0; nt < 4; ++nt) {
#pragma unroll
    for (int r = 0; r < 8; ++r) {
      int row = row0 + r + 8 * half;
      int col = nt * 16 + l15;
      float v = acc[nt][r] + bias[col];
      out[(size_t)row * DIM + col] = leaky(v);
    }
  }
}

// ---------------------------------------------------------------------------
// Row l2norm over 64-wide rows: dst[r] = (a[r] (+ b[r])) / max(||.||, 1e-12)
// One wave per row, 2 floats per lane.
// ---------------------------------------------------------------------------
__global__ void k_row_l2norm(float* __restrict__ dst, const float* __restrict__ a,
                             const float* __restrict__ pb, int rows) {
  int wave = (int)((blockIdx.x * blockDim.x + threadIdx.x) >> 5);
  int lane = threadIdx.x & 31;
  if (wave >= rows) return;
  size_t base = (size_t)wave * DIM + lane * 2;
  float v0 = a[base], v1 = a[base + 1];
  if (pb) { v0 += pb[base]; v1 += pb[base + 1]; }
  float ss = v0 * v0 + v1 * v1;
#pragma unroll
  for (int m = 16; m >= 1; m >>= 1) ss += __shfl_xor(ss, m, 32);
  float inv = 1.0f / fmaxf(sqrtf(ss), 1e-12f);
  dst[base] = v0 * inv;
  dst[base + 1] = v1 * inv;
}

// ---------------------------------------------------------------------------
// GAT phase 1: per-edge logit = dot(x[dst], x[src]); atomic segment max over dst.
// 8 lanes per edge. Edges e >= E1 are the reversed copies (src/dst swapped).
// dstLimit: routing passes only need segments with dst < NUM_USER.
// ---------------------------------------------------------------------------
__global__ void k_edge_logit(const int* __restrict__ src, const int* __restrict__ dst,
                             const float* __restrict__ x, float* __restrict__ alpha,
                             int* __restrict__ segmax, int E, int E1, int dstLimit) {
  long long t = (long long)blockIdx.x * blockDim.x + threadIdx.x;
  int e = (int)(t >> 3), sub = (int)(t & 7);
  if (e >= E) return;
  int s, d;
  if (e < E1) { s = src[e]; d = dst[e]; } else { s = dst[e - E1]; d = src[e - E1]; }
  if (d >= dstLimit) return;
  const float* xs = x + (size_t)s * DIM;
  const float* xd = x + (size_t)d * DIM;
  float p = 0.f;
#pragma unroll
  for (int i = 0; i < 8; ++i) p += xs[i * 8 + sub] * xd[i * 8 + sub];
  p += __shfl_xor(p, 4, 32);
  p += __shfl_xor(p, 2, 32);
  p += __shfl_xor(p, 1, 32);
  if (sub == 0) {
    alpha[e] = p;
    atomicMax(&segmax[d], f2ord(p));
  }
}

// GAT phase 2: alpha = exp(logit - segmax[dst]); atomic segment sum.
__global__ void k_edge_expsum(const int* __restrict__ src, const int* __restrict__ dst,
                              float* __restrict__ alpha, const int* __restrict__ segmax,
                              float* __restrict__ segsum, int E, int E1, int dstLimit) {
  int e = blockIdx.x * blockDim.x + threadIdx.x;
  if (e >= E) return;
  int d = (e < E1) ? dst[e] : src[e - E1];
  if (d >= dstLimit) return;
  float mf = ord2f(segmax[d]);
  float ev = __expf(alpha[e] - mf);
  alpha[e] = ev;
  atomicAdd(&segsum[d], ev);
}

// GAT phase 3: normalize alpha; xh[dst] += x[src] * alpha. 8 lanes per edge.
__global__ void k_edge_scatter(const int* __restrict__ src, const int* __restrict__ dst,
                               const float* __restrict__ x, float* __restrict__ alpha,
                               const float* __restrict__ segsum, float* __restrict__ xh,
                               int E, int E1, int dstLimit) {
  long long t = (long long)blockIdx.x * blockDim.x + threadIdx.x;
  int e = (int)(t >> 3), sub = (int)(t & 7);
  if (e >= E) return;
  int s, d;
  if (e < E1) { s = src[e]; d = dst[e]; } else { s = dst[e - E1]; d = src[e - E1]; }
  if (d >= dstLimit) return;
  float a = alpha[e] / (segsum[d] + 1e-16f);  // all lanes read before any write
  const float* xs = x + (size_t)s * DIM;
  float* xo = xh + (size_t)d * DIM;
#pragma unroll
  for (int i = 0; i < 8; ++i) atomicAdd(&xo[i * 8 + sub], xs[i * 8 + sub] * a);
  if (sub == 0) alpha[e] = a;  // store normalized attention (returned by gat_conv)
}

// rep = x + leaky_relu(xh), written into out columns [colOff, colOff+64)
__global__ void k_finalize_rep(float* __restrict__ out, const float* __restrict__ x,
                               const float* __restrict__ xh, int colOff) {
  long long t = (long long)blockIdx.x * blockDim.x + threadIdx.x;
  if (t >= (long long)NNODE * DIM) return;
  int n = (int)(t >> 6), i = (int)(t & 63);
  out[(size_t)n * (3 * DIM) + colOff + i] = x[t] + leaky(xh[t]);
}

// weight[e] = relu(max(alpha_v[e]*conf[s][0], alpha_t[e]*conf[s][1])), s = src of ei2 edge
__global__ void k_weight(const int* __restrict__ src, const int* __restrict__ dst,
                         const float* __restrict__ av, const float* __restrict__ at,
                         const float* __restrict__ conf, float* __restrict__ w,
                         int E2, int E1) {
  int e = blockIdx.x * blockDim.x + threadIdx.x;
  if (e >= E2) return;
  int s = (e < E1) ? src[e] : dst[e - E1];
  float wv = av[e] * conf[(size_t)s * 2 + 0];
  float wt = at[e] * conf[(size_t)s * 2 + 1];
  w[e] = fmaxf(fmaxf(wv, wt), 0.f);
}

// SAGE scatter: acc[dst] += xin[src] * w[e]; cnt[dst] += 1 (first layer only)
__global__ void k_sage_scatter(const int* __restrict__ src, const int* __restrict__ dst,
                               const float* __restrict__ xin, const float* __restrict__ w,
                               float* __restrict__ acc, float* __restrict__ cnt,
                               int E2, int E1, int addCnt) {
  long long t = (long long)blockIdx.x * blockDim.x + threadIdx.x;
  int e = (int)(t >> 3), sub = (int)(t & 7);
  if (e >= E2) return;
  int s, d;
  if (e < E1) { s = src[e]; d = dst[e]; } else { s = dst[e - E1]; d = src[e - E1]; }
  float we = w[e];
  const float* xs = xin + (size_t)s * DIM;
  float* xo = acc + (size_t)d * DIM;
#pragma unroll
  for (int i = 0; i < 8; ++i) atomicAdd(&xo[i * 8 + sub], xs[i * 8 + sub] * we);
  if (addCnt && sub == 0) atomicAdd(&cnt[d], 1.0f);
}

// h = leaky_relu(acc / max(cnt, 1))
__global__ void k_sage_finish(float* __restrict__ h, const float* __restrict__ acc,
                              const float* __restrict__ cnt) {
  long long t = (long long)blockIdx.x * blockDim.x + threadIdx.x;
  if (t >= (long long)NNODE * DIM) return;
  int n = (int)(t >> 6);
  float v = acc[t] / fmaxf(cnt[n], 1.0f);
  h[t] = leaky(v);
}

// out[:, 0:64] = xn + h1 + leaky_relu(acc / max(cnt, 1))
__global__ void k_egcn_final(float* __restrict__ out, const float* __restrict__ xn,
                             const float* __restrict__ h1, const float* __restrict__ acc,
                             const float* __restrict__ cnt) {
  long long t = (long long)blockIdx.x * blockDim.x + threadIdx.x;
  if (t >= (long long)NNODE * DIM) return;
  int n = (int)(t >> 6), i = (int)(t & 63);
  float v = acc[t] / fmaxf(cnt[n], 1.0f);
  out[(size_t)n * (3 * DIM) + i] = xn[t] + h1[t] + leaky(v);
}

// ---------------------------------------------------------------------------
static inline int nb(long long threads) { return (int)((threads + 255) / 256); }

static void run_gat(const int* src, const int* dst, const float* x, float* xh,
                    float* alpha, int* segmax, float* segsum,
                    int E, int E1, int dstLimit, hipStream_t s) {
  hipMemsetAsync(segmax, 0x80, NNODE * sizeof(int), s);  // ~ -3.4e38 after ord2f
  hipMemsetAsync(segsum, 0, NNODE * sizeof(float), s);
  hipMemsetAsync(xh, 0, (size_t)dstLimit * DIM * sizeof(float), s);
  long long tE8 = (long long)E * 8;
  k_edge_logit<<<nb(tE8), 256, 0, s>>>(src, dst, x, alpha, segmax, E, E1, dstLimit);
  k_edge_expsum<<<nb(E), 256, 0, s>>>(src, dst, alpha, segmax, segsum, E, E1, dstLimit);
  k_edge_scatter<<<nb(tE8), 256, 0, s>>>(src, dst, x, alpha, segsum, xh, E, E1, dstLimit);
}

static void run_cgcn(const float* feat, int K, const float* Wm, const float* bias,
                     const float* pref, const int* src, const int* dst, int E1,
                     float* x, float* xh, float* alpha, int* segmax, float* segsum,
                     float* out, int colOff, hipStream_t s) {
  // features = l2norm(leaky_relu(feat @ W + b)) -> x rows [NUM_USER, NNODE)
  float* feats = x + (size_t)NUM_USER * DIM;
  k_proj_wmma<<<nb((long long)(NUM_ITEM / 16) * 32), 256, 0, s>>>(feat, Wm, bias, feats,
                                                                  NUM_ITEM, K);
  k_row_l2norm<<<nb((long long)NUM_ITEM * 32), 256, 0, s>>>(feats, feats, nullptr, NUM_ITEM);
  // pref = l2norm(pref) -> x rows [0, NUM_USER)
  k_row_l2norm<<<nb((long long)NUM_USER * 32), 256, 0, s>>>(x, pref, nullptr, NUM_USER);
  // 3 routing iterations (only user-dst segments matter)
  for (int r = 0; r < 3; ++r) {
    run_gat(src, dst, x, xh, alpha, segmax, segsum, E1, E1, NUM_USER, s);
    k_row_l2norm<<<nb((long long)NUM_USER * 32), 256, 0, s>>>(x, x, xh, NUM_USER);
  }
  // final GAT on doubled edge list (2M edges); keep normalized alpha
  run_gat(src, dst, x, xh, alpha, segmax, segsum, 2 * E1, E1, NNODE, s);
  k_finalize_rep<<<nb((long long)NNODE * DIM), 256, 0, s>>>(out, x, xh, colOff);
}

extern "C" void kernel_launch(void* const* d_in, const int* in_sizes, int n_in,
                              void* d_out, int out_size, void* d_ws, size_t ws_size,
                              hipStream_t stream) {
  const int* ei       = (const int*)d_in[0];
  const float* v_feat = (const float*)d_in[1];
  const float* t_feat = (const float*)d_in[2];
  const float* pref_v = (const float*)d_in[3];
  const float* pref_t = (const float*)d_in[4];
  const float* W_v    = (const float*)d_in[5];
  const float* b_v    = (const float*)d_in[6];
  const float* W_t    = (const float*)d_in[7];
  const float* b_t    = (const float*)d_in[8];
  const float* id_emb = (const float*)d_in[9];
  const float* conf   = (const float*)d_in[10];
  float* out = (float*)d_out;

  const int E1 = in_sizes[0] / 2;           // 1,000,000 directed edges
  const int E2 = 2 * E1;                    // with reversed copies
  const int Kv = in_sizes[1] / NUM_ITEM;    // 2048
  const int Kt = in_sizes[2] / NUM_ITEM;    // 384
  const int* src = ei;                      // edge_index[0]
  const int* dst = ei + E1;                 // edge_index[1]

  const size_t NN64 = (size_t)NNODE * DIM;
  float* x       = (float*)d_ws;            // node features / l2norm(id_emb)
  float* xh      = x + NN64;                // GAT / SAGE aggregation target
  float* h1      = xh + NN64;               // egcn layer-1 output
  float* alpha_v = h1 + NN64;               // [E2]
  float* alpha_t = alpha_v + (size_t)E2;    // [E2]
  float* weight  = alpha_t + (size_t)E2;    // [E2]
  int*   segmax  = (int*)(weight + (size_t)E2);
  float* segsum  = (float*)(segmax + NNODE);
  float* cnt     = segsum + NNODE;

  // ---- content branches ----
  run_cgcn(v_feat, Kv, W_v, b_v, pref_v, src, dst, E1, x, xh, alpha_v, segmax, segsum,
           out, /*colOff=*/DIM, stream);
  run_cgcn(t_feat, Kt, W_t, b_t, pref_t, src, dst, E1, x, xh, alpha_t, segmax, segsum,
           out, /*colOff=*/2 * DIM, stream);

  // ---- edge weight pruning ----
  k_weight<<<nb(E2), 256, 0, stream>>>(src, dst, alpha_v, alpha_t, conf, weight, E2, E1);

  // ---- egcn: x + h1 + h2 with weighted SAGE-mean ----
  k_row_l2norm<<<nb((long long)NNODE * 32), 256, 0, stream>>>(x, id_emb, nullptr, NNODE);
  hipMemsetAsync(xh, 0, NN64 * sizeof(float), stream);
  hipMemsetAsync(cnt, 0, NNODE * sizeof(float), stream);
  k_sage_scatter<<<nb((long long)E2 * 8), 256, 0, stream>>>(src, dst, x, weight, xh, cnt,
                                                            E2, E1, /*addCnt=*/1);
  k_sage_finish<<<nb((long long)NNODE * DIM), 256, 0, stream>>>(h1, xh, cnt);
  hipMemsetAsync(xh, 0, NN64 * sizeof(float), stream);
  k_sage_scatter<<<nb((long long)E2 * 8), 256, 0, stream>>>(src, dst, h1, weight, xh, cnt,
                                                            E2, E1, /*addCnt=*/0);
  k_egcn_final<<<nb((long long)NNODE * DIM), 256, 0, stream>>>(out, x, h1, xh, cnt);
}